// JacobiMachine_39238821216639
// MI455X (gfx1250) — compile-verified
//
#include <hip/hip_runtime.h>

// Output = 100 masked-Jacobi sweeps of X (1024x1024, f32) with Mask1.
// Temporal blocking: 96x96 LDS region per block (64x64 output + 16 halo),
// up to 16 steps per launch -> 7 launches total. Stencil computed per-wave
// as T*X + X*T via V_WMMA_F32_16X16X4_F32 (tridiagonal T), halo edges fixed
// with masked VALU adds. Staging via async global->LDS loads when available.

#define NSIDE  1024
#define OUTT   64
#define HALO   16
#define REG    96            // OUTT + 2*HALO, = 6 x 16 tiles
#define PITCH  100           // LDS row pitch (floats): float4-aligned, conflict-free strides
#define TPB    256
#define NWAVE  8
#define NTILES 36            // 6x6 tiles of 16x16
#define LDS_FLOATS (3 * REG * PITCH)
#define LDS_BYTES  (LDS_FLOATS * 4)

typedef __attribute__((ext_vector_type(2))) float v2f;
typedef __attribute__((ext_vector_type(4))) float v4f;
typedef __attribute__((ext_vector_type(8))) float v8f;

#if defined(__gfx1250__) && __has_builtin(__builtin_amdgcn_wmma_f32_16x16x4_f32)
#define HAVE_WMMA_F32 1
#endif

#if defined(__gfx1250__) && \
    __has_builtin(__builtin_amdgcn_global_load_async_to_lds_b128) && \
    __has_builtin(__builtin_amdgcn_s_wait_asynccnt)
#define HAVE_ASYNC 1
// Builtin signature (from clang diagnostic): (as1 v4i*, as3 v4i*, imm offset, imm cpol)
typedef int b128_t __attribute__((vector_size(16)));
typedef __attribute__((address_space(1))) b128_t* g_b128_p;
typedef __attribute__((address_space(3))) b128_t* l_b128_p;
#endif

__device__ __forceinline__ int iclamp(int v, int lo, int hi) {
    return v < lo ? lo : (v > hi ? hi : v);
}

__global__ __launch_bounds__(TPB) void jm_jacobi_stage(
    const float* __restrict__ src, const float* __restrict__ mask,
    float* __restrict__ dst, int steps)
{
    extern __shared__ float lds[];
    float* Xa = lds;                    // ping
    float* Xb = lds + REG * PITCH;      // pong
    float* Mm = lds + 2 * REG * PITCH;  // mask tile

    const int tid = threadIdx.x;
    const int bx  = blockIdx.x & 15;
    const int by  = blockIdx.x >> 4;
    const int gx0 = bx * OUTT - HALO;   // global col of region col 0 (multiple of 16 offset)
    const int gy0 = by * OUTT - HALO;

    // ---------------- stage x + mask into LDS (out-of-domain -> 0) ----------------
    // float4 chunks: each chunk is fully inside or fully outside the domain
    // (gx0 is a multiple of 16, NSIDE a multiple of 4).
    for (int idx = tid; idx < REG * (REG / 4); idx += TPB) {
        const int r    = idx / (REG / 4);
        const int c4   = (idx - r * (REG / 4)) * 4;
        const int grow = gy0 + r;
        const int gcol = gx0 + c4;
        float* lx = &Xa[r * PITCH + c4];
        float* lm = &Mm[r * PITCH + c4];
        if ((unsigned)grow < NSIDE && (unsigned)gcol < NSIDE) {
            const float* gp = src  + (size_t)grow * NSIDE + gcol;
            const float* mp = mask + (size_t)grow * NSIDE + gcol;
#if defined(HAVE_ASYNC)
            __builtin_amdgcn_global_load_async_to_lds_b128((g_b128_p)gp, (l_b128_p)lx, 0, 0);
            __builtin_amdgcn_global_load_async_to_lds_b128((g_b128_p)mp, (l_b128_p)lm, 0, 0);
#else
            *(v4f*)lx = *(const v4f*)gp;
            *(v4f*)lm = *(const v4f*)mp;
#endif
        } else {
            v4f z = {0.f, 0.f, 0.f, 0.f};
            *(v4f*)lx = z;
            *(v4f*)lm = z;
        }
    }
#if defined(HAVE_ASYNC)
    __builtin_amdgcn_s_wait_asynccnt(0);
#endif
    __syncthreads();

    // ---------------- per-lane identities ----------------
    const int lane = tid & 31;
    const int wave = tid >> 5;
    const int ncol = lane & 15;     // N index (column within tile)
    const int hi   = lane >> 4;     // 0: lanes 0-15, 1: lanes 16-31

#if defined(HAVE_WMMA_F32)
    // Tridiagonal T (0.25 on sub/super diagonals), chunked along K in the
    // f32 16x16x4 A-layout. T is symmetric, so the same per-lane values serve
    // as the B-layout chunks too (A: lane m=ncol, K=4k+2*hi+j ; B: K rows, n=ncol).
    v2f tC[4];
#pragma unroll
    for (int k = 0; k < 4; ++k) {
#pragma unroll
        for (int j = 0; j < 2; ++j) {
            const int kap = 4 * k + 2 * hi + j;
            const int d   = kap - ncol;
            tC[k][j] = (d == 1 || d == -1) ? 0.25f : 0.0f;
        }
    }
#endif

    float* cur = Xa;
    float* nxt = Xb;
    for (int t = 0; t < steps; ++t) {
        for (int tile = wave; tile < NTILES; tile += NWAVE) {   // wave-uniform loop
            const int tr = (tile / 6) * 16;   // region row of tile origin
            const int tc = (tile % 6) * 16;   // region col of tile origin
            v8f conv;

#if defined(HAVE_WMMA_F32)
            v8f acc = {0.f, 0.f, 0.f, 0.f, 0.f, 0.f, 0.f, 0.f};
            // vertical neighbors: acc += T(16x4) * X(4x16), K-chunks k=0..3
#pragma unroll
            for (int k = 0; k < 4; ++k) {
                v2f bX;
                const int rB = tr + 4 * k + 2 * hi;
                bX[0] = cur[(rB + 0) * PITCH + tc + ncol];
                bX[1] = cur[(rB + 1) * PITCH + tc + ncol];
                acc = __builtin_amdgcn_wmma_f32_16x16x4_f32(
                        false, tC[k], false, bX, (short)0, acc, false, false);
            }
            // horizontal neighbors: acc += X(16x4) * T(4x16), K-chunks k=0..3
#pragma unroll
            for (int k = 0; k < 4; ++k) {
                const v2f aX = *(const v2f*)&cur[(tr + ncol) * PITCH + tc + 4 * k + 2 * hi];
                acc = __builtin_amdgcn_wmma_f32_16x16x4_f32(
                        false, aX, false, tC[k], (short)0, acc, false, false);
            }
            conv = acc;
            // inter-tile halo fixes (rows 0/15, cols 0/15 of the tile)
            {
                int rh = hi ? (tr + 16) : (tr - 1);
                rh = iclamp(rh, 0, REG - 1);
                const float hv = 0.25f * cur[rh * PITCH + tc + ncol];
                if (hi == 0) conv[0] += hv; else conv[7] += hv;
                if (ncol == 0 || ncol == 15) {
                    int cc = (ncol == 0) ? (tc - 1) : (tc + 16);
                    cc = iclamp(cc, 0, REG - 1);
#pragma unroll
                    for (int i = 0; i < 8; ++i)
                        conv[i] += 0.25f * cur[(tr + i + 8 * hi) * PITCH + cc];
                }
            }
#else
            // scalar fallback: direct 4-neighbor gather (clamped at region edge)
#pragma unroll
            for (int i = 0; i < 8; ++i) {
                const int r = tr + i + 8 * hi;
                const int c = tc + ncol;
                const int rm = r > 0 ? r - 1 : 0;
                const int rp = r < REG - 1 ? r + 1 : REG - 1;
                const int cm = c > 0 ? c - 1 : 0;
                const int cp = c < REG - 1 ? c + 1 : REG - 1;
                conv[i] = 0.25f * (cur[rm * PITCH + c] + cur[rp * PITCH + c] +
                                   cur[r * PITCH + cm] + cur[r * PITCH + cp]);
            }
#endif
            // masked blend, keep out-of-domain cells pinned to 0 (SAME zero padding)
#pragma unroll
            for (int i = 0; i < 8; ++i) {
                const int r = tr + i + 8 * hi;
                const int c = tc + ncol;
                const float cen = cur[r * PITCH + c];
                const float m   = Mm [r * PITCH + c];
                const int grow = gy0 + r, gcol = gx0 + c;
                const bool inside = (unsigned)grow < NSIDE && (unsigned)gcol < NSIDE;
                const float v = fmaf(m, conv[i] - cen, cen);   // m*conv + (1-m)*cen
                nxt[r * PITCH + c] = inside ? v : 0.0f;
            }
        }
        __syncthreads();
        float* tswap = cur; cur = nxt; nxt = tswap;
    }

    // ---------------- write central OUTT x OUTT to global ----------------
    for (int idx = tid; idx < OUTT * OUTT; idx += TPB) {
        const int r = idx >> 6;
        const int c = idx & 63;
        dst[(size_t)(by * OUTT + r) * NSIDE + bx * OUTT + c] =
            cur[(HALO + r) * PITCH + HALO + c];
    }
}

extern "C" void kernel_launch(void* const* d_in, const int* in_sizes, int n_in,
                              void* d_out, int out_size, void* d_ws, size_t ws_size,
                              hipStream_t stream) {
    const float* X  = (const float*)d_in[0];   // 1x1x1024x1024
    const float* M1 = (const float*)d_in[1];   // Mask1, 1024x1024
    float* out  = (float*)d_out;
    float* bufA = (float*)d_ws;
    float* bufB = bufA + (size_t)NSIDE * NSIDE;

    // 100 steps = 16*6 + 4, halo supports up to 16 steps per launch
    const int chunks[7] = {16, 16, 16, 16, 16, 16, 4};
    const float* srcp = X;
    for (int i = 0; i < 7; ++i) {
        float* dstp = (i == 6) ? out : ((i & 1) ? bufB : bufA);
        jm_jacobi_stage<<<dim3(256), dim3(TPB), LDS_BYTES, stream>>>(srcp, M1, dstp, chunks[i]);
        srcp = dstp;
    }
    (void)in_sizes; (void)n_in; (void)out_size; (void)ws_size;
}